// GNN_69355131896366
// MI455X (gfx1250) — compile-verified
//
#include <hip/hip_runtime.h>
#include <cmath>
#include <cstddef>

typedef float v2f __attribute__((ext_vector_type(2)));
typedef float v8f __attribute__((ext_vector_type(8)));

static constexpr int kNA  = 200000;
static constexpr int kNT  = 300000;
static constexpr int kNG  = 10000;
static constexpr int kE   = 500000;
static constexpr int kH   = 128;
static constexpr int kOUT = 64;

// ---------------------------------------------------------------- kernels

__global__ void k_fill(float* __restrict__ p, float v, size_t n) {
  size_t i  = (size_t)blockIdx.x * blockDim.x + threadIdx.x;
  size_t st = (size_t)gridDim.x * blockDim.x;
  for (; i < n; i += st) p[i] = v;
}

// C[M,N] (+)= A[M,K] @ B[K,N] (+ bias[N]); row-major, leading dims lda/ldb/ldc.
// One wave32 computes one 16x16 tile via V_WMMA_F32_16X16X4_F32.
// Requires M%16==0, N%16==0, K%4==0 (true for every call below).
// A layout: lanes 0-15 hold rows M=0..15 with {K=k,k+1}; lanes 16-31 same rows {K=k+2,k+3}.
// B layout: lanes 0-15 hold cols N=0..15 with {K=k,k+1}; lanes 16-31 {K=k+2,k+3}.
// C layout: VGPR r -> row r (lanes 0-15) / row r+8 (lanes 16-31), col = lane&15.
__global__ __launch_bounds__(128) void k_gemm_wmma(
    const float* __restrict__ A, int lda,
    const float* __restrict__ B, int ldb,
    float* __restrict__ C, int ldc,
    const float* __restrict__ bias,
    int M, int K, int N, int accumulate)
{
  int lane = threadIdx.x & 31;
  int wave = threadIdx.x >> 5;
  int row0 = blockIdx.x * 16;
  int col0 = (blockIdx.y * 4 + wave) * 16;
  if (row0 >= M || col0 >= N) return;       // wave-uniform exit: EXEC stays full for WMMA
  int half = lane >> 4;
  int l    = lane & 15;
  const float* Arow = A + (size_t)(row0 + l) * lda;
  const float* Bcol = B + (size_t)(col0 + l);
  union { v8f v; float f[8]; } acc;
  acc.v = (v8f){0.f,0.f,0.f,0.f,0.f,0.f,0.f,0.f};
  for (int k = 0; k < K; k += 4) {
    int ka = k + half * 2;
    v2f av, bv;
    av.x = Arow[ka];
    av.y = Arow[ka + 1];
    bv.x = Bcol[(size_t)ka * ldb];
    bv.y = Bcol[(size_t)(ka + 1) * ldb];
    acc.v = __builtin_amdgcn_wmma_f32_16x16x4_f32(
        false, av, false, bv, (short)0, acc.v, false, false);
  }
  int col = col0 + l;
  float badd = bias ? bias[col] : 0.0f;
#pragma unroll
  for (int r = 0; r < 8; ++r) {
    int row = row0 + half * 8 + r;
    size_t idx = (size_t)row * ldc + col;
    float v = acc.f[r] + badd;
    if (accumulate) C[idx] += v; else C[idx] = v;
  }
}

// y[i] = dot(X[i, 0:K], v[0:K]); one wave per row.
__global__ void k_rowdot(const float* __restrict__ X, const float* __restrict__ v,
                         float* __restrict__ y, int M, int K) {
  int gid  = blockIdx.x * blockDim.x + threadIdx.x;
  int w    = gid >> 5;
  int lane = gid & 31;
  if (w >= M) return;
  const float* xr = X + (size_t)w * K;
  float s = 0.f;
  for (int c = lane; c < K; c += 32) s += xr[c] * v[c];
  for (int off = 16; off; off >>= 1) s += __shfl_xor(s, off, 32);
  if (lane == 0) y[w] = s;
}

__device__ __forceinline__ void atomicMaxF(float* addr, float val) {
  int* ai = (int*)addr;
  int old = __float_as_int(*addr);
  while (__int_as_float(old) < val) {
    int prev = atomicCAS(ai, old, __float_as_int(val));
    if (prev == old) break;
    old = prev;
  }
}

__device__ __forceinline__ float leaky(float e) { return e > 0.f ? e : 0.2f * e; }

// item i < E: edge; item in [E, E+Nd): self-loop (src=dst=i-E)
__global__ void k_gat_max(const int* __restrict__ src, const int* __restrict__ dst,
                          int E, int Nd, const float* __restrict__ es,
                          const float* __restrict__ ed, float* __restrict__ m) {
  int i = blockIdx.x * blockDim.x + threadIdx.x;
  if (i >= E + Nd) return;
  int s, d;
  if (i < E) { s = src[i]; d = dst[i]; } else { s = d = i - E; }
  atomicMaxF(&m[d], leaky(es[s] + ed[d]));
}

__global__ void k_gat_expsum(const int* __restrict__ src, const int* __restrict__ dst,
                             int E, int Nd, const float* __restrict__ es,
                             const float* __restrict__ ed, const float* __restrict__ m,
                             float* __restrict__ den) {
  int i = blockIdx.x * blockDim.x + threadIdx.x;
  if (i >= E + Nd) return;
  int s, d;
  if (i < E) { s = src[i]; d = dst[i]; } else { s = d = i - E; }
  float e = leaky(es[s] + ed[d]);
  atomicAdd(&den[d], __expf(e - m[d]));
}

// one wave per item; lanes stride the C channels
__global__ void k_gat_scatter(const int* __restrict__ src, const int* __restrict__ dst,
                              int E, int Nd, const float* __restrict__ es,
                              const float* __restrict__ ed, const float* __restrict__ m,
                              const float* __restrict__ den, const float* __restrict__ hs,
                              float* __restrict__ acc, int C) {
  int gid  = blockIdx.x * blockDim.x + threadIdx.x;
  int w    = gid >> 5;
  int lane = gid & 31;
  if (w >= E + Nd) return;
  int s, d;
  if (w < E) { s = src[w]; d = dst[w]; } else { s = d = w - E; }
  float e = leaky(es[s] + ed[d]);
  float alpha = __expf(e - m[d]) / den[d];
  const float* hrow = hs + (size_t)s * C;
  float* arow = acc + (size_t)d * C;
  for (int c = lane; c < C; c += 32) atomicAdd(&arow[c], hrow[c] * alpha);
}

__global__ void k_sage_scatter(const int* __restrict__ src, const int* __restrict__ dst,
                               int E, const float* __restrict__ xs,
                               float* __restrict__ sum, float* __restrict__ cnt, int Ds) {
  int gid  = blockIdx.x * blockDim.x + threadIdx.x;
  int w    = gid >> 5;
  int lane = gid & 31;
  if (w >= E) return;
  int s = src[w], d = dst[w];
  const float* xr = xs + (size_t)s * Ds;
  float* sr = sum + (size_t)d * Ds;
  for (int c = lane; c < Ds; c += 32) atomicAdd(&sr[c], xr[c]);
  if (lane == 0) atomicAdd(&cnt[d], 1.0f);
}

__global__ void k_sage_mean(float* __restrict__ sum, const float* __restrict__ cnt,
                            int Nd, int Ds) {
  size_t i = (size_t)blockIdx.x * blockDim.x + threadIdx.x;
  if (i >= (size_t)Nd * Ds) return;
  sum[i] /= fmaxf(cnt[i / Ds], 1.0f);
}

// out[i,c] = (acc[i,c] + extra[c]) * scale   (extra may be null)
__global__ void k_scale_bias(const float* __restrict__ acc, float* __restrict__ out,
                             const float* __restrict__ extra, float scale,
                             int Nn, int C) {
  size_t i = (size_t)blockIdx.x * blockDim.x + threadIdx.x;
  if (i >= (size_t)Nn * C) return;
  float e = extra ? extra[i % C] : 0.0f;
  out[i] = (acc[i] + e) * scale;
}

__global__ void k_sum5(const float* a, const float* b, const float* c,
                       const float* d, const float* e, float* o, int n) {
  int i = blockIdx.x * blockDim.x + threadIdx.x;
  if (i < n) o[i] = a[i] + b[i] + c[i] + d[i] + e[i];
}

__global__ void k_l2norm(float* __restrict__ out, int M, int C) {
  int gid  = blockIdx.x * blockDim.x + threadIdx.x;
  int w    = gid >> 5;
  int lane = gid & 31;
  if (w >= M) return;
  float* row = out + (size_t)w * C;
  float s = 0.f;
  for (int c = lane; c < C; c += 32) { float v = row[c]; s += v * v; }
  for (int off = 16; off; off >>= 1) s += __shfl_xor(s, off, 32);
  float inv = 1.0f / fmaxf(sqrtf(s), 1e-12f);
  for (int c = lane; c < C; c += 32) row[c] *= inv;
}

// ---------------------------------------------------------------- host

// top-level input indices (setup_inputs insertion order)
enum {
  IN_XA = 0, IN_XT, IN_XG,
  IN_E_COLLAB, IN_E_LASTFM, IN_E_LINKED, IN_E_MUSICAL, IN_E_PERSONAL,
  IN_E_HTA, IN_E_HTT, IN_E_TA, IN_E_TT, IN_E_WB, IN_E_WI,
  IN_W1, IN_W2, IN_B1, IN_B2,
  IN_CONV1 = 18
};
static constexpr int CONV_LEAVES = 43;          // 5 GAT*5 + 6 SAGE*3
static constexpr int IN_CONV2 = IN_CONV1 + CONV_LEAVES;
// relation leaf offsets within a conv block (jax sorted-key flatten)
static constexpr int R_COLLAB   = 0;   // gat:  Wd, Ws, ad, as_, b
static constexpr int R_HTA      = 5;   // sage: Wl, Wr, bl
static constexpr int R_HTT      = 8;
static constexpr int R_LASTFM   = 11;
static constexpr int R_LINKED   = 16;
static constexpr int R_MUSICAL  = 21;
static constexpr int R_PERSONAL = 26;
static constexpr int R_TA       = 29;
static constexpr int R_TT       = 32;
static constexpr int R_WB       = 35;
static constexpr int R_WI       = 38;

extern "C" void kernel_launch(void* const* d_in, const int* in_sizes, int n_in,
                              void* d_out, int out_size, void* d_ws, size_t ws_size,
                              hipStream_t stream) {
  (void)in_sizes; (void)n_in; (void)out_size; (void)ws_size;

  auto F = [&](int i) -> const float* { return (const float*)d_in[i]; };
  auto Ei = [&](int i) -> const int*  { return (const int*)d_in[i]; };

  // -------- workspace carve-out (floats) --------
  float* ws = (float*)d_ws;
  size_t o = 0;
  float* ACC_A = ws + o; o += (size_t)kNA * kH;      // artist accumulator (layer1 & layer2)
  float* A1    = ws + o; o += (size_t)kNA * kH;      // layer-1 artist output
  float* ACC_G = ws + o; o += (size_t)kNG * kH;      // tag accumulator -> g1 (in place)
  float* ES    = ws + o; o += kNA;
  float* EDv   = ws + o; o += kNA;
  float* MX    = ws + o; o += kNA;
  float* DEN   = ws + o; o += kNA;
  float* CNT   = ws + o; o += kNT;
  float* WV    = ws + o; o += kH;
  float* BSUM  = ws + o; o += kH;
  // aliased region: HS + MEAN + ACC_T (= exactly NA*512 floats) reused as MLP hidden H1
  float* HS    = ws + o;
  float* MEAN  = HS + (size_t)kNA * kH;              // max Nd*Ds = NT*128
  float* ACC_T = MEAN + (size_t)kNT * kH;            // track accumulator -> t1 (in place)
  float* H1    = HS;                                 // [NA, 512] after convs are done

  auto nblk = [](size_t threads) { return dim3((unsigned)((threads + 255) / 256)); };

  auto fill = [&](float* p, size_t n, float v) {
    size_t b = (n + 255) / 256; if (b > 262144) b = 262144;
    k_fill<<<dim3((unsigned)b), 256, 0, stream>>>(p, v, n);
  };
  auto gemm = [&](const float* A, int lda, const float* B, int ldb, float* C, int ldc,
                  const float* bias, int M, int K, int N, int accum) {
    dim3 g(M / 16, (N / 16 + 3) / 4);
    k_gemm_wmma<<<g, 128, 0, stream>>>(A, lda, B, ldb, C, ldc, bias, M, K, N, accum);
  };
  auto rowdot = [&](const float* X, const float* v, float* y, int M, int K) {
    k_rowdot<<<nblk((size_t)M * 32), 256, 0, stream>>>(X, v, y, M, K);
  };

  // GAT artist->artist: accumulate segment-softmax messages into ACC_A (bias folded later)
  auto run_gat = [&](const float* x, int base, int rel, int edgeIdx) {
    const float* Wd  = F(base + rel + 0);
    const float* Ws  = F(base + rel + 1);
    const float* ad  = F(base + rel + 2);
    const float* as_ = F(base + rel + 3);
    const int* e = Ei(edgeIdx);
    gemm(x, kH, Ws, kH, HS, kH, nullptr, kNA, kH, kH, 0);  // hs = x @ Ws
    rowdot(Wd, ad, WV, kH, kH);                            // wv = Wd @ ad
    rowdot(x, WV, EDv, kNA, kH);                           // ed = x @ wv  (== (x@Wd)@ad)
    rowdot(HS, as_, ES, kNA, kH);                          // es = hs @ as_
    fill(MX, kNA, -INFINITY);
    fill(DEN, kNA, 0.0f);
    int n = kE + kNA;                                      // edges + self-loops
    k_gat_max   <<<nblk(n), 256, 0, stream>>>(e, e + kE, kE, kNA, ES, EDv, MX);
    k_gat_expsum<<<nblk(n), 256, 0, stream>>>(e, e + kE, kE, kNA, ES, EDv, MX, DEN);
    k_gat_scatter<<<nblk((size_t)n * 32), 256, 0, stream>>>(
        e, e + kE, kE, kNA, ES, EDv, MX, DEN, HS, ACC_A, kH);
  };

  // SAGE: acc += mean(x_src over in-edges) @ Wl + bl + x_dst @ Wr
  auto run_sage = [&](const float* xs, int Ds, const float* xd, int Dd, int Nd,
                      int edgeIdx, int base, int rel, float* acc) {
    const float* Wl = F(base + rel + 0);
    const float* Wr = F(base + rel + 1);
    const float* bl = F(base + rel + 2);
    const int* e = Ei(edgeIdx);
    fill(MEAN, (size_t)Nd * Ds, 0.0f);
    fill(CNT, Nd, 0.0f);
    k_sage_scatter<<<nblk((size_t)kE * 32), 256, 0, stream>>>(e, e + kE, kE, xs, MEAN, CNT, Ds);
    k_sage_mean<<<nblk((size_t)Nd * Ds), 256, 0, stream>>>(MEAN, CNT, Nd, Ds);
    gemm(MEAN, Ds, Wl, kH, acc, kH, bl, Nd, Ds, kH, 1);
    gemm(xd, Dd, Wr, kH, acc, kH, nullptr, Nd, Dd, kH, 1);
  };

  auto gat_bias_sum = [&](int base) {
    k_sum5<<<1, 128, 0, stream>>>(F(base + R_COLLAB + 4), F(base + R_LASTFM + 4),
                                  F(base + R_LINKED + 4), F(base + R_MUSICAL + 4),
                                  F(base + R_PERSONAL + 4), BSUM, kH);
  };

  const float* XA = F(IN_XA);
  const float* XT = F(IN_XT);
  const float* XG = F(IN_XG);

  // ================= layer 1 =================
  fill(ACC_A, (size_t)kNA * kH, 0.0f);
  fill(ACC_G, (size_t)kNG * kH, 0.0f);
  fill(ACC_T, (size_t)kNT * kH, 0.0f);

  run_gat(XA, IN_CONV1, R_COLLAB,   IN_E_COLLAB);
  run_gat(XA, IN_CONV1, R_LASTFM,   IN_E_LASTFM);
  run_gat(XA, IN_CONV1, R_LINKED,   IN_E_LINKED);
  run_gat(XA, IN_CONV1, R_MUSICAL,  IN_E_MUSICAL);
  run_gat(XA, IN_CONV1, R_PERSONAL, IN_E_PERSONAL);
  run_sage(XG, 64,  XA, kH, kNA, IN_E_TA, IN_CONV1, R_TA, ACC_A);   // tag  -> artist
  run_sage(XT, kH,  XA, kH, kNA, IN_E_WB, IN_CONV1, R_WB, ACC_A);   // track-> artist
  gat_bias_sum(IN_CONV1);
  k_scale_bias<<<nblk((size_t)kNA * kH), 256, 0, stream>>>(ACC_A, A1, BSUM, 1.0f / 7.0f, kNA, kH);

  run_sage(XA, kH, XG, 64, kNG, IN_E_HTA, IN_CONV1, R_HTA, ACC_G);  // artist-> tag
  run_sage(XT, kH, XG, 64, kNG, IN_E_HTT, IN_CONV1, R_HTT, ACC_G);  // track -> tag
  k_scale_bias<<<nblk((size_t)kNG * kH), 256, 0, stream>>>(ACC_G, ACC_G, nullptr, 0.5f, kNG, kH);

  run_sage(XG, 64, XT, kH, kNT, IN_E_TT, IN_CONV1, R_TT, ACC_T);    // tag   -> track
  run_sage(XA, kH, XT, kH, kNT, IN_E_WI, IN_CONV1, R_WI, ACC_T);    // artist-> track
  k_scale_bias<<<nblk((size_t)kNT * kH), 256, 0, stream>>>(ACC_T, ACC_T, nullptr, 0.5f, kNT, kH);

  // ================= layer 2 (only artist output feeds the head) =================
  fill(ACC_A, (size_t)kNA * kH, 0.0f);
  run_gat(A1, IN_CONV2, R_COLLAB,   IN_E_COLLAB);
  run_gat(A1, IN_CONV2, R_LASTFM,   IN_E_LASTFM);
  run_gat(A1, IN_CONV2, R_LINKED,   IN_E_LINKED);
  run_gat(A1, IN_CONV2, R_MUSICAL,  IN_E_MUSICAL);
  run_gat(A1, IN_CONV2, R_PERSONAL, IN_E_PERSONAL);
  run_sage(ACC_G, kH, A1, kH, kNA, IN_E_TA, IN_CONV2, R_TA, ACC_A); // g1 -> artist
  run_sage(ACC_T, kH, A1, kH, kNA, IN_E_WB, IN_CONV2, R_WB, ACC_A); // t1 -> artist
  gat_bias_sum(IN_CONV2);
  k_scale_bias<<<nblk((size_t)kNA * kH), 256, 0, stream>>>(ACC_A, ACC_A, BSUM, 1.0f / 7.0f, kNA, kH);

  // ================= MLP head: [a1 | a2] @ W1 + b1 -> @ W2 + b2 -> L2 norm ==========
  // H1 aliases HS/MEAN/ACC_T which are all dead now.
  const float* W1 = F(IN_W1);   // [256, 512]
  const float* W2 = F(IN_W2);   // [512, 64]
  const float* b1 = F(IN_B1);
  const float* b2 = F(IN_B2);
  float* OUT = (float*)d_out;

  fill(H1, (size_t)kNA * 512, 0.0f);
  gemm(A1,    kH, W1,            512, H1, 512, b1,      kNA, kH, 512, 1);  // a1 @ W1[:128]
  gemm(ACC_A, kH, W1 + 128 * 512, 512, H1, 512, nullptr, kNA, kH, 512, 1); // a2 @ W1[128:]
  gemm(H1, 512, W2, kOUT, OUT, kOUT, b2, kNA, 512, kOUT, 0);
  k_l2norm<<<nblk((size_t)kNA * 32), 256, 0, stream>>>(OUT, kNA, kOUT);
}